// PaddedSHCSA_13443247636578
// MI455X (gfx1250) — compile-verified
//
#include <hip/hip_runtime.h>

// ---------------------------------------------------------------------------
// PaddedSHCSA for MI455X (gfx1250): fused QKV GEMM + masked causal attention.
// All matmuls use V_WMMA_F32_16X16X4_F32 (native f32 WMMA, wave32).
// Round 2: B-tiles stored transposed in LDS so both A and B fragments are
// single aligned 8-byte contiguous LDS loads (kills the v_mov shuffle
// overhead seen in round 1), plus global_prefetch of the next K-chunk.
// ---------------------------------------------------------------------------

typedef float v2f __attribute__((ext_vector_type(2)));
typedef float v8f __attribute__((ext_vector_type(8)));

#define SEQ   4096
#define FEAT  2048
#define QKVF  6144
#define SCALE 0.022097086912079608f   // 1/sqrt(2048)

#define LDA 34    // A-tile [64 rows][32 K], padded even stride -> b64 frags
#define LDBT 34   // B-tile stored transposed [128 n][32 K], even stride

__device__ __forceinline__ v8f wmma4(v2f a, v2f b, v8f c) {
  // 8 args: (neg_a, A, neg_b, B, c_mod, C, reuse_a, reuse_b)
  return __builtin_amdgcn_wmma_f32_16x16x4_f32(
      false, a, false, b, (short)0, c, false, false);
}

// ---------------------------------------------------------------------------
// Kernel 1: qkv = x @ W + b   -> writes q [S,F], kT [F,S] (transposed), v [S,F]
// Grid: (QKVF/128, SEQ/64), block 256 (8 wave32).
// ---------------------------------------------------------------------------
__global__ __launch_bounds__(256)
void qkv_gemm(const float* __restrict__ x, const float* __restrict__ W,
              const float* __restrict__ bias,
              float* __restrict__ q, float* __restrict__ kT,
              float* __restrict__ v) {
  __shared__ float As[64 * LDA];
  __shared__ float Bs[128 * LDBT];
  const int t    = threadIdx.x;
  const int lane = t & 31;
  const int w    = t >> 5;
  const int mt   = w & 3;            // wave's 16-row tile within 64 rows
  const int ntb  = (w >> 2) * 4;     // wave's first 16-col tile within 128 cols
  const int m_base = blockIdx.y * 64;
  const int n_base = blockIdx.x * 128;

  const int acol = t & 31,  arb = (t >> 5) * 8;
  const int bcol = t & 127, brb = (t >> 7) * 16;

  v8f c[4] = {};

  for (int kc = 0; kc < FEAT; kc += 32) {
    // stage: A = x[m_base..+64, kc..+32] row-major; B = W[kc..+32, n..+128]
    // stored transposed in LDS as Bs[n][k].
#pragma unroll
    for (int r = 0; r < 8; ++r)
      As[(arb + r) * LDA + acol] =
          x[(size_t)(m_base + arb + r) * FEAT + kc + acol];
#pragma unroll
    for (int r = 0; r < 16; ++r)
      Bs[bcol * LDBT + brb + r] =
          W[(size_t)(kc + brb + r) * QKVF + n_base + bcol];
    if (kc + 32 < FEAT) {   // prefetch next chunk (global_prefetch_b8)
      __builtin_prefetch(&x[(size_t)(m_base + arb) * FEAT + kc + 32 + acol], 0, 0);
      __builtin_prefetch(&W[(size_t)(kc + 32 + brb) * QKVF + n_base + bcol], 0, 0);
    }
    __syncthreads();

    const int hi2 = (lane >> 4) << 1;   // lanes 16-31 hold K+2 / K+3
    const int arow = (mt * 16 + (lane & 15)) * LDA;
    const int bn0  = (ntb * 16 + (lane & 15)) * LDBT;
#pragma unroll
    for (int kk = 0; kk < 32; kk += 4) {
      const v2f a = *(const v2f*)&As[arow + kk + hi2];
#pragma unroll
      for (int j = 0; j < 4; ++j) {
        const v2f b = *(const v2f*)&Bs[bn0 + j * 16 * LDBT + kk + hi2];
        c[j] = wmma4(a, b, c[j]);
      }
    }
    __syncthreads();
  }

  // epilogue: add bias, scatter into q / kT(transposed) / v
  const int hi = lane >> 4;
#pragma unroll
  for (int j = 0; j < 4; ++j) {
    const int gcol = n_base + (ntb + j) * 16 + (lane & 15);
    const float bv = bias[gcol];
#pragma unroll
    for (int e = 0; e < 8; ++e) {
      const int grow = m_base + mt * 16 + hi * 8 + e;
      const float val = c[j][e] + bv;
      if (gcol < FEAT) {
        q[(size_t)grow * FEAT + gcol] = val;
      } else if (gcol < 2 * FEAT) {
        kT[(size_t)(gcol - FEAT) * SEQ + grow] = val;   // store K transposed
      } else {
        v[(size_t)grow * FEAT + (gcol - 2 * FEAT)] = val;
      }
    }
  }
}

// ---------------------------------------------------------------------------
// Kernel 2: att_raw = (q @ kT) * SCALE  (lower-triangular blocks only)
// Grid: (SEQ/128, SEQ/64), block 256.
// ---------------------------------------------------------------------------
__global__ __launch_bounds__(256)
void scores_gemm(const float* __restrict__ q, const float* __restrict__ kT,
                 float* __restrict__ att) {
  const int m_base = blockIdx.y * 64;
  const int n_base = blockIdx.x * 128;
  if (n_base > m_base + 63) return;   // fully above diagonal: never read

  __shared__ float As[64 * LDA];
  __shared__ float Bs[128 * LDBT];
  const int t    = threadIdx.x;
  const int lane = t & 31;
  const int w    = t >> 5;
  const int mt   = w & 3;
  const int ntb  = (w >> 2) * 4;

  const int acol = t & 31,  arb = (t >> 5) * 8;
  const int bcol = t & 127, brb = (t >> 7) * 16;

  v8f c[4] = {};

  for (int kc = 0; kc < FEAT; kc += 32) {
#pragma unroll
    for (int r = 0; r < 8; ++r)
      As[(arb + r) * LDA + acol] =
          q[(size_t)(m_base + arb + r) * FEAT + kc + acol];
#pragma unroll
    for (int r = 0; r < 16; ++r)
      Bs[bcol * LDBT + brb + r] =
          kT[(size_t)(kc + brb + r) * SEQ + n_base + bcol];
    if (kc + 32 < FEAT) {
      __builtin_prefetch(&q[(size_t)(m_base + arb) * FEAT + kc + 32 + acol], 0, 0);
      __builtin_prefetch(&kT[(size_t)(kc + 32 + brb) * SEQ + n_base + bcol], 0, 0);
    }
    __syncthreads();

    const int hi2 = (lane >> 4) << 1;
    const int arow = (mt * 16 + (lane & 15)) * LDA;
    const int bn0  = (ntb * 16 + (lane & 15)) * LDBT;
#pragma unroll
    for (int kk = 0; kk < 32; kk += 4) {
      const v2f a = *(const v2f*)&As[arow + kk + hi2];
#pragma unroll
      for (int j = 0; j < 4; ++j) {
        const v2f b = *(const v2f*)&Bs[bn0 + j * 16 * LDBT + kk + hi2];
        c[j] = wmma4(a, b, c[j]);
      }
    }
    __syncthreads();
  }

  const int hi = lane >> 4;
#pragma unroll
  for (int j = 0; j < 4; ++j) {
    const int gcol = n_base + (ntb + j) * 16 + (lane & 15);
#pragma unroll
    for (int e = 0; e < 8; ++e) {
      const int grow = m_base + mt * 16 + hi * 8 + e;
      att[(size_t)grow * SEQ + gcol] = c[j][e] * SCALE;
    }
  }
}

// ---------------------------------------------------------------------------
// Kernel 3: masked, numerically-stable row softmax in place.
// Valid columns: n_padd <= j <= i. Writes exact zeros elsewhere so the
// final GEMM can clip its K-loop at the diagonal. Grid: SEQ, block 256.
// ---------------------------------------------------------------------------
__global__ __launch_bounds__(256)
void row_softmax(float* __restrict__ att, const int* __restrict__ npp) {
  const int i  = blockIdx.x;
  const int np = *npp;
  if (i < np) return;   // padded rows: never read by out_gemm (zeroed there)

  __shared__ float red[256];
  const int t = threadIdx.x;
  float* row = att + (size_t)i * SEQ;

  float lm = -__builtin_inff();
  for (int j = np + t; j <= i; j += 256) lm = fmaxf(lm, row[j]);
  red[t] = lm;
  __syncthreads();
  for (int s = 128; s > 0; s >>= 1) {
    if (t < s) red[t] = fmaxf(red[t], red[t + s]);
    __syncthreads();
  }
  const float m = red[0];
  __syncthreads();

  float ls = 0.0f;
  for (int j = np + t; j <= i; j += 256) ls += __expf(row[j] - m);
  red[t] = ls;
  __syncthreads();
  for (int s = 128; s > 0; s >>= 1) {
    if (t < s) red[t] += red[t + s];
    __syncthreads();
  }
  const float inv = 1.0f / red[0];

  for (int j = t; j < SEQ; j += 256) {
    float val = 0.0f;
    if (j >= np && j <= i) val = __expf(row[j] - m) * inv;
    row[j] = val;
  }
}

// ---------------------------------------------------------------------------
// Kernel 4: out = att @ v, K-loop clipped at the diagonal; padded rows -> 0.
// Grid: (FEAT/128, SEQ/64), block 256.
// ---------------------------------------------------------------------------
__global__ __launch_bounds__(256)
void out_gemm(const float* __restrict__ att, const float* __restrict__ v,
              float* __restrict__ out, const int* __restrict__ npp) {
  const int np     = *npp;
  const int m_base = blockIdx.y * 64;
  const int n_base = blockIdx.x * 128;
  const int t      = threadIdx.x;

  if (m_base + 64 <= np) {   // whole block is padding: store zeros
    for (int idx = t; idx < 64 * 128; idx += 256) {
      const int r = idx >> 7, cc = idx & 127;
      out[(size_t)(m_base + r) * FEAT + n_base + cc] = 0.0f;
    }
    return;
  }

  __shared__ float As[64 * LDA];
  __shared__ float Bs[128 * LDBT];
  const int lane = t & 31;
  const int w    = t >> 5;
  const int mt   = w & 3;
  const int ntb  = (w >> 2) * 4;

  const int acol = t & 31,  arb = (t >> 5) * 8;
  const int bcol = t & 127, brb = (t >> 7) * 16;

  v8f c[4] = {};
  const int kend = m_base + 64;   // att cols > row are exact zeros

  for (int kc = 0; kc < kend; kc += 32) {
#pragma unroll
    for (int r = 0; r < 8; ++r)
      As[(arb + r) * LDA + acol] =
          att[(size_t)(m_base + arb + r) * SEQ + kc + acol];
#pragma unroll
    for (int r = 0; r < 16; ++r)
      Bs[bcol * LDBT + brb + r] =
          v[(size_t)(kc + brb + r) * FEAT + n_base + bcol];
    if (kc + 32 < kend) {
      __builtin_prefetch(&att[(size_t)(m_base + arb) * SEQ + kc + 32 + acol], 0, 0);
      __builtin_prefetch(&v[(size_t)(kc + 32 + brb) * FEAT + n_base + bcol], 0, 0);
    }
    __syncthreads();

    const int hi2 = (lane >> 4) << 1;
    const int arow = (mt * 16 + (lane & 15)) * LDA;
    const int bn0  = (ntb * 16 + (lane & 15)) * LDBT;
#pragma unroll
    for (int kk = 0; kk < 32; kk += 4) {
      const v2f a = *(const v2f*)&As[arow + kk + hi2];
#pragma unroll
      for (int j = 0; j < 4; ++j) {
        const v2f b = *(const v2f*)&Bs[bn0 + j * 16 * LDBT + kk + hi2];
        c[j] = wmma4(a, b, c[j]);
      }
    }
    __syncthreads();
  }

  const int hi = lane >> 4;
#pragma unroll
  for (int j = 0; j < 4; ++j) {
    const int gcol = n_base + (ntb + j) * 16 + (lane & 15);
#pragma unroll
    for (int e = 0; e < 8; ++e) {
      const int grow = m_base + mt * 16 + hi * 8 + e;
      out[(size_t)grow * FEAT + gcol] = (grow < np) ? 0.0f : c[j][e];
    }
  }
}

// ---------------------------------------------------------------------------
// Launcher. Workspace layout (floats):
//   q:  [0, 8M)      kT: [8M, 16M)    v: [16M, 24M)    att: [24M, 40M)
// Total 160 MB.
// ---------------------------------------------------------------------------
extern "C" void kernel_launch(void* const* d_in, const int* in_sizes, int n_in,
                              void* d_out, int out_size, void* d_ws,
                              size_t ws_size, hipStream_t stream) {
  const float* x    = (const float*)d_in[0];
  const float* W    = (const float*)d_in[1];
  const float* bias = (const float*)d_in[2];
  const int*   npp  = (const int*)d_in[3];
  float*       out  = (float*)d_out;

  float* q   = (float*)d_ws;
  float* kT  = q  + (size_t)SEQ * FEAT;
  float* v   = kT + (size_t)SEQ * FEAT;
  float* att = v  + (size_t)SEQ * FEAT;

  qkv_gemm<<<dim3(QKVF / 128, SEQ / 64), 256, 0, stream>>>(x, W, bias, q, kT, v);
  scores_gemm<<<dim3(SEQ / 128, SEQ / 64), 256, 0, stream>>>(q, kT, att);
  row_softmax<<<dim3(SEQ), 256, 0, stream>>>(att, npp);
  out_gemm<<<dim3(FEAT / 128, SEQ / 64), 256, 0, stream>>>(att, v, out, npp);
}